// ProjectionLayer_72035191489074
// MI455X (gfx1250) — compile-verified
//
#include <hip/hip_runtime.h>
#include <math.h>

#define POSE_SCALING 0.001f
#define BB 32
#define HH 256
#define WW 512

typedef __attribute__((ext_vector_type(2))) float v2f;
typedef __attribute__((ext_vector_type(8))) float v8f;

// Select element (row,col) of a 3x3 (register-resident) matrix, 0 if outside.
// Fully unrolled constant indices -> pure v_cndmask chain, no scratch.
__device__ __forceinline__ float sel33(const float (&r)[9], int row, int col) {
    float v = 0.0f;
#pragma unroll
    for (int rr = 0; rr < 3; ++rr)
#pragma unroll
        for (int cc = 0; cc < 3; ++cc)
            v = (row == rr && col == cc) ? r[rr * 3 + cc] : v;
    return v;
}

// One wave (32 lanes) per batch. Computes M = K @ R @ K_inv with two
// V_WMMA_F32_16X16X4_F32 ops (3x3 embedded in 16x4 / 4x16 operands),
// plus T = K @ (t * 0.001). Writes 12 floats per batch to MT.
__global__ void pose_compose_kernel(const float* __restrict__ pose,
                                    const float* __restrict__ K,
                                    const float* __restrict__ Kinv,
                                    float* __restrict__ MT) {
    const int b    = blockIdx.x;
    const int lane = threadIdx.x;      // 0..31, full wave (EXEC all ones)
    const int m    = lane & 15;
    const int hi   = lane >> 4;        // 0: cols/rows {0,1}, 1: cols/rows {2,3}

    // ---- per-batch scalars (uniform across the wave) ----
    const float tx = pose[b * 6 + 0] * POSE_SCALING;
    const float ty = pose[b * 6 + 1] * POSE_SCALING;
    const float tz = pose[b * 6 + 2] * POSE_SCALING;
    const float rx = pose[b * 6 + 3] * POSE_SCALING;
    const float ry = pose[b * 6 + 4] * POSE_SCALING;
    const float rz = pose[b * 6 + 5] * POSE_SCALING;

    const float cxa = cosf(rx), sxa = sinf(rx);
    const float cya = cosf(ry), sya = sinf(ry);
    const float cza = cosf(rz), sza = sinf(rz);

    // R = Rz @ Ry @ Rx
    float R[9];
    R[0] = cza * cya;
    R[1] = -sza * cxa + cza * sya * sxa;
    R[2] = sza * sxa + cza * sya * cxa;
    R[3] = sza * cya;
    R[4] = cza * cxa + sza * sya * sxa;
    R[5] = -cza * sxa + sza * sya * cxa;
    R[6] = -sya;
    R[7] = cya * sxa;
    R[8] = cya * cxa;

    float Kv[9], Ki[9];
#pragma unroll
    for (int i = 0; i < 9; ++i) { Kv[i] = K[i]; Ki[i] = Kinv[i]; }

    // ---- WMMA #1: T1 = R @ Kinv ----
    // A (16x4, f32): lane<16 holds row m cols {0,1}; lane>=16 holds cols {2,3}.
    v2f aR;
    aR.x = sel33(R, m, hi ? 2 : 0);
    aR.y = sel33(R, m, hi ? 3 : 1);
    // B (4x16, f32): VGPR0 = rows {0 | 2} striped over lane groups, VGPR1 = rows {1 | 3}.
    v2f bKi;
    bKi.x = sel33(Ki, hi ? 2 : 0, m);
    bKi.y = sel33(Ki, hi ? 3 : 1, m);

    v8f czero = {};
    v8f d1 = __builtin_amdgcn_wmma_f32_16x16x4_f32(false, aR, false, bKi,
                                                   (short)0, czero, false, false);
    // D layout: VGPR r, lanes 0-15 = row r col lane (rows >=3 are zero here).

    // Repack T1 (rows 0..2) from D layout into B layout for the next WMMA.
    const float t1r2 = __shfl(d1[2], m, 32);   // row 2, col m -> needed in lanes 16-31
    v2f bT1;
    bT1.x = hi ? t1r2 : d1[0];                 // rows {0 | 2}
    bT1.y = hi ? 0.0f : d1[1];                 // rows {1 | 3(=0)}

    // ---- WMMA #2: M = K @ T1 ----
    v2f aK;
    aK.x = sel33(Kv, m, hi ? 2 : 0);
    aK.y = sel33(Kv, m, hi ? 3 : 1);
    v8f d2 = __builtin_amdgcn_wmma_f32_16x16x4_f32(false, aK, false, bT1,
                                                   (short)0, czero, false, false);

    // ---- write out M (rows 0..2 x cols 0..2) and T = K @ t ----
    float* o = MT + b * 12;
    if (lane < 3) {
        o[0 * 3 + lane] = d2[0];   // row 0, col = lane
        o[1 * 3 + lane] = d2[1];   // row 1
        o[2 * 3 + lane] = d2[2];   // row 2
        if (lane == 0) {
            o[9]  = Kv[0] * tx + Kv[1] * ty + Kv[2] * tz;
            o[10] = Kv[3] * tx + Kv[4] * ty + Kv[5] * tz;
            o[11] = Kv[6] * tx + Kv[7] * ty + Kv[8] * tz;
        }
    }
}

// One thread per output pixel. Batch/row index derived from blockIdx only,
// so the 12 per-batch constants become scalar (s_load) reads.
__global__ void __launch_bounds__(256)
warp_bilinear_kernel(const float* __restrict__ src,
                     const float* __restrict__ depth,
                     const float* __restrict__ MT,
                     float* __restrict__ out) {
    const int gbase = blockIdx.x * 256;                 // 256 consecutive pixels, same b, same row
    const int b     = gbase >> 17;                      // / (H*W) = /131072
    const int rem   = gbase & 131071;
    const int py    = rem >> 9;                         // / W = /512
    const int px    = (rem & 511) + threadIdx.x;        // rem&511 is 0 or 256
    const int idx   = gbase + threadIdx.x;

    // Scalar (wave-uniform) matrix + translation
    const float* mt = MT + b * 12;
    const float m0 = mt[0], m1 = mt[1], m2 = mt[2];
    const float m3 = mt[3], m4 = mt[4], m5 = mt[5];
    const float m6 = mt[6], m7 = mt[7], m8 = mt[8];
    const float Tx = mt[9], Ty = mt[10], Tz = mt[11];

    const float d  = depth[idx];                        // coalesced b32
    const float fx = (float)px;
    const float fy = (float)py;

    // proj = d * (M @ [x,y,1]) + T
    const float zx = fmaf(m0, fx, fmaf(m1, fy, m2));
    const float zy = fmaf(m3, fx, fmaf(m4, fy, m5));
    const float zz = fmaf(m6, fx, fmaf(m7, fy, m8));
    const float X  = fmaf(d, zx, Tx);
    const float Y  = fmaf(d, zy, Ty);
    const float Z  = fmaf(d, zz, Tz) + 1e-10f;

    const float inv = 1.0f / Z;
    const float xs  = X * inv;
    const float ys  = Y * inv;

    // Bilinear sample with border clamp + zero-outside mask (matches reference)
    const float x0f = floorf(xs);
    const float y0f = floorf(ys);
    const float wx1 = xs - x0f, wy1 = ys - y0f;
    const float wx0 = 1.0f - wx1, wy0 = 1.0f - wy1;

    const int x0 = (int)fminf(fmaxf(x0f,        0.0f), (float)(WW - 1));
    const int x1 = (int)fminf(fmaxf(x0f + 1.0f, 0.0f), (float)(WW - 1));
    const int y0 = (int)fminf(fmaxf(y0f,        0.0f), (float)(HH - 1));
    const int y1 = (int)fminf(fmaxf(y0f + 1.0f, 0.0f), (float)(HH - 1));

    const bool valid = (xs >= 0.0f) & (xs <= (float)(WW - 1)) &
                       (ys >= 0.0f) & (ys <= (float)(HH - 1));
    const float vm = valid ? 1.0f : 0.0f;

    const float w00 = wx0 * wy0 * vm;
    const float w01 = wx1 * wy0 * vm;
    const float w10 = wx0 * wy1 * vm;
    const float w11 = wx1 * wy1 * vm;

    const float* base = src + (size_t)b * (HH * WW * 3);
    const float* p00 = base + (size_t)(y0 * WW + x0) * 3;
    const float* p01 = base + (size_t)(y0 * WW + x1) * 3;
    const float* p10 = base + (size_t)(y1 * WW + x0) * 3;
    const float* p11 = base + (size_t)(y1 * WW + x1) * 3;

    float r0 = p00[0] * w00, r1 = p00[1] * w00, r2 = p00[2] * w00;  // b96-mergeable taps
    r0 = fmaf(p01[0], w01, r0); r1 = fmaf(p01[1], w01, r1); r2 = fmaf(p01[2], w01, r2);
    r0 = fmaf(p10[0], w10, r0); r1 = fmaf(p10[1], w10, r1); r2 = fmaf(p10[2], w10, r2);
    r0 = fmaf(p11[0], w11, r0); r1 = fmaf(p11[1], w11, r1); r2 = fmaf(p11[2], w11, r2);

    float* o = out + (size_t)idx * 3;   // consecutive 12B per lane -> coalesced
    o[0] = r0; o[1] = r1; o[2] = r2;
}

extern "C" void kernel_launch(void* const* d_in, const int* in_sizes, int n_in,
                              void* d_out, int out_size, void* d_ws, size_t ws_size,
                              hipStream_t stream) {
    const float* src   = (const float*)d_in[0];  // [32,256,512,3]
    const float* depth = (const float*)d_in[1];  // [32,256,512,1]
    const float* pose  = (const float*)d_in[2];  // [32,6]
    const float* K     = (const float*)d_in[3];  // [3,3]
    const float* Kinv  = (const float*)d_in[4];  // [3,3]
    float* out = (float*)d_out;
    float* MT  = (float*)d_ws;                   // 32 * 12 floats

    pose_compose_kernel<<<BB, 32, 0, stream>>>(pose, K, Kinv, MT);

    const int npix = BB * HH * WW;               // 4,194,304
    warp_bilinear_kernel<<<npix / 256, 256, 0, stream>>>(src, depth, MT, out);
}